// Seq2Seq_86328842649851
// MI455X (gfx1250) — compile-verified
//
#include <hip/hip_runtime.h>
#include <hip/hip_bf16.h>

typedef __attribute__((ext_vector_type(2))) float v2f;
typedef __attribute__((ext_vector_type(8))) float v8f;

#define WMMA_F32(a, b, c) \
    __builtin_amdgcn_wmma_f32_16x16x4_f32(false, (a), false, (b), (short)0, (c), false, false)

static __device__ __forceinline__ float sigmoidf_(float x) {
    return 1.0f / (1.0f + __expf(-x));
}
static __device__ __forceinline__ v2f ld2(const float* p) {
    return *(const v2f*)p;       // 8B-aligned global_load_b64 / ds_load_b64
}

// ---------------------------------------------------------------------------
// Embedding gather (optionally BOS-shifted for the decoder, optional ReLU)
// ---------------------------------------------------------------------------
__global__ void gather_emb(const int* __restrict__ toks, const float* __restrict__ emb,
                           float* __restrict__ out, int T, int B, int H,
                           int relu, int shift_bos) {
    size_t i = (size_t)blockIdx.x * blockDim.x + threadIdx.x;
    size_t total = (size_t)T * B * H;
    if (i >= total) return;
    int hc = (int)(i % H);
    int tb = (int)(i / H);
    int b = tb % B, t = tb / B;
    int tok = shift_bos ? ((t == 0) ? 1 : toks[(t - 1) * B + b]) : toks[tb];
    float v = emb[(size_t)tok * H + hc];
    if (relu) v = fmaxf(v, 0.0f);
    out[i] = v;
}

// ---------------------------------------------------------------------------
// C[M,N] = A[M,K] @ W[N,K]^T + bias.   fp32 WMMA 16x16x4.
// Block = 256 threads = 8 waves covering a 32(M) x 256(N) region.
// Each wave owns a 16x64 strip (4 C tiles): one A fragment feeds 4 WMMAs.
// Software-pipelined, unroll disabled so the steady-state loop keeps the
// 5 prefetch loads of step i+1 in flight across the 4 WMMAs of step i
// (in-order VMEM completion -> pre-WMMA wait is loadcnt<=5, never 0).
// Out-of-range columns: B row pointer clamped (safe load); garbage stays in
// columns we never store (WMMA columns are independent) -> no masking.
// ---------------------------------------------------------------------------
__global__ __launch_bounds__(256) void gemm_nt(const float* __restrict__ A, int lda,
                                               const float* __restrict__ W,
                                               const float* __restrict__ bias,
                                               float* __restrict__ C, int ldc,
                                               int M, int N, int K) {
    const int wave = threadIdx.x >> 5;
    const int lane = threadIdx.x & 31;
    const int half = lane >> 4;     // k-half select
    const int lan  = lane & 15;     // m (for A) / n (for B) within tile
    const int m0 = blockIdx.y * 32 + (wave >> 2) * 16;
    const int n0 = blockIdx.x * 256 + (wave & 3) * 64;
    if (m0 >= M) return;            // wave-uniform

    const float* ap = A + (size_t)(m0 + lan) * lda;
    const int nc0 = n0 + 0 * 16 + lan;
    const int nc1 = n0 + 1 * 16 + lan;
    const int nc2 = n0 + 2 * 16 + lan;
    const int nc3 = n0 + 3 * 16 + lan;
    const float* wp0 = W + (size_t)(nc0 < N ? nc0 : (N - 1)) * K;
    const float* wp1 = W + (size_t)(nc1 < N ? nc1 : (N - 1)) * K;
    const float* wp2 = W + (size_t)(nc2 < N ? nc2 : (N - 1)) * K;
    const float* wp3 = W + (size_t)(nc3 < N ? nc3 : (N - 1)) * K;

    v8f c0 = {}, c1 = {}, c2 = {}, c3 = {};

    int kk = half * 2;
    v2f a_c  = ld2(ap  + kk);
    v2f b0_c = ld2(wp0 + kk);
    v2f b1_c = ld2(wp1 + kk);
    v2f b2_c = ld2(wp2 + kk);
    v2f b3_c = ld2(wp3 + kk);
#pragma clang loop unroll(disable)
    for (int k0 = 4; k0 < K; k0 += 4) {
        int kn = k0 + half * 2;
        v2f a_n  = ld2(ap  + kn);
        v2f b0_n = ld2(wp0 + kn);
        v2f b1_n = ld2(wp1 + kn);
        v2f b2_n = ld2(wp2 + kn);
        v2f b3_n = ld2(wp3 + kn);
        c0 = WMMA_F32(a_c, b0_c, c0);
        c1 = WMMA_F32(a_c, b1_c, c1);
        c2 = WMMA_F32(a_c, b2_c, c2);
        c3 = WMMA_F32(a_c, b3_c, c3);
        a_c = a_n; b0_c = b0_n; b1_c = b1_n; b2_c = b2_n; b3_c = b3_n;
    }
    c0 = WMMA_F32(a_c, b0_c, c0);
    c1 = WMMA_F32(a_c, b1_c, c1);
    c2 = WMMA_F32(a_c, b2_c, c2);
    c3 = WMMA_F32(a_c, b3_c, c3);

#pragma unroll
    for (int q = 0; q < 4; ++q) {
        v8f cc = (q == 0) ? c0 : (q == 1) ? c1 : (q == 2) ? c2 : c3;
        int n = (q == 0) ? nc0 : (q == 1) ? nc1 : (q == 2) ? nc2 : nc3;
        if (n < N) {
            float bb = bias ? bias[n] : 0.0f;
#pragma unroll
            for (int r = 0; r < 8; ++r) {
                int m = m0 + r + half * 8;
                if (m < M) C[(size_t)m * ldc + n] = cc[r] + bb;
            }
        }
    }
}

// ---------------------------------------------------------------------------
// Sequential GRU over T steps. B=32, H=256 hardcoded.
// One 1024-thread block (32 waves). h lives in LDS, row stride padded to 260
// floats (260 mod 64 = 4) so the 16 lanes of an A-fragment load hit 16
// distinct banks instead of all colliding on one.
// Wave w: m-tile = w&1, j-tile = w>>1; computes its r,z,n gate tiles
// (gh = h @ Whh^T) in registers (1 A fragment feeds 3 WMMAs, double-buffered,
// unroll disabled for uniform pipelining), then does the elementwise gate
// update in-place on LDS h.
// gi = x @ Wih^T + bih is precomputed (parallel over time) in global memory.
// ---------------------------------------------------------------------------
#define HP 260   // padded LDS row stride (floats)

struct GruArgs {
    const float* gi;    // (T, 32, 768)
    const float* Whh;   // (768, 256)
    const float* bhh;   // (768)
    const float* h0;    // (32, 256) or null (zeros)
    float* outs;        // (T, 32, 256) or null
    float* hfin;        // final hidden, row stride hfin_stride, or null
    int hfin_stride;
    int T;
};

__global__ __launch_bounds__(1024) void gru_seq(GruArgs ga, GruArgs gb) {
    GruArgs g = (blockIdx.x == 0) ? ga : gb;
    __shared__ float h[32 * HP];
    const int tid = threadIdx.x;
    for (int i = tid; i < 32 * 256; i += 1024)
        h[(i >> 8) * HP + (i & 255)] = g.h0 ? g.h0[i] : 0.0f;
    __syncthreads();

    const int wave = tid >> 5, lane = tid & 31;
    const int half = lane >> 4, lan = lane & 15;
    const int m0 = (wave & 1) * 16;
    const int j  = (wave >> 1) * 16 + lan;     // output column 0..255

    const float* Wr = g.Whh + (size_t)j * 256;
    const float* Wz = g.Whh + (size_t)(256 + j) * 256;
    const float* Wn = g.Whh + (size_t)(512 + j) * 256;
    const float br_b = g.bhh[j], bz_b = g.bhh[256 + j], bn_b = g.bhh[512 + j];

    for (int t = 0; t < g.T; ++t) {
        v8f cr = {}, cz = {}, cn = {};
        const float* hrow = &h[(m0 + lan) * HP];

        int kk = half * 2;
        v2f a_c  = ld2(hrow + kk);
        v2f br_c = ld2(Wr + kk);
        v2f bz_c = ld2(Wz + kk);
        v2f bn_c = ld2(Wn + kk);
#pragma clang loop unroll(disable)
        for (int k0 = 4; k0 < 256; k0 += 4) {
            int kn = k0 + half * 2;
            v2f a_n  = ld2(hrow + kn);
            v2f br_n = ld2(Wr + kn);
            v2f bz_n = ld2(Wz + kn);
            v2f bn_n = ld2(Wn + kn);
            cr = WMMA_F32(a_c, br_c, cr);
            cz = WMMA_F32(a_c, bz_c, cz);
            cn = WMMA_F32(a_c, bn_c, cn);
            a_c = a_n; br_c = br_n; bz_c = bz_n; bn_c = bn_n;
        }
        cr = WMMA_F32(a_c, br_c, cr);
        cz = WMMA_F32(a_c, bz_c, cz);
        cn = WMMA_F32(a_c, bn_c, cn);

        float hn[8];
        const float* gib = g.gi + (size_t)t * 32 * 768;
#pragma unroll
        for (int r = 0; r < 8; ++r) {
            int m = m0 + r + half * 8;                  // batch row
            const float* gi = gib + (size_t)m * 768;
            float ghr = cr[r] + br_b;
            float ghz = cz[r] + bz_b;
            float ghn = cn[r] + bn_b;
            float rr = sigmoidf_(gi[j] + ghr);
            float zz = sigmoidf_(gi[256 + j] + ghz);
            float nn = tanhf(gi[512 + j] + rr * ghn);
            hn[r] = (1.0f - zz) * nn + zz * h[m * HP + j];
        }
        __syncthreads();   // all waves done reading h
#pragma unroll
        for (int r = 0; r < 8; ++r) {
            int m = m0 + r + half * 8;
            h[m * HP + j] = hn[r];
            if (g.outs) g.outs[((size_t)t * 32 + m) * 256 + j] = hn[r];
        }
        __syncthreads();   // h fully updated before next step's matmul
    }
    if (g.hfin) {
        for (int i = tid; i < 32 * 256; i += 1024)
            g.hfin[(size_t)(i >> 8) * g.hfin_stride + (i & 255)] =
                h[(i >> 8) * HP + (i & 255)];
    }
}

// ---------------------------------------------------------------------------
// Per-row online log-sum-exp + target NLL.  One block per (t,b) row.
// ---------------------------------------------------------------------------
__global__ __launch_bounds__(256) void row_nll(const float* __restrict__ logits,
                                               const int* __restrict__ target,
                                               int V, float* __restrict__ nll) {
    const int row = blockIdx.x;
    const float* x = logits + (size_t)row * V;
    float mx = -3.402823466e38f, s = 0.0f;
    for (int v = threadIdx.x; v < V; v += 256) {
        float xv = x[v];
        float nm = fmaxf(mx, xv);
        s = s * __expf(mx - nm) + __expf(xv - nm);
        mx = nm;
    }
    __shared__ float smx[256], ss[256];
    smx[threadIdx.x] = mx; ss[threadIdx.x] = s;
    __syncthreads();
    for (int off = 128; off > 0; off >>= 1) {
        if (threadIdx.x < off) {
            float m1 = smx[threadIdx.x], s1 = ss[threadIdx.x];
            float m2 = smx[threadIdx.x + off], s2 = ss[threadIdx.x + off];
            float nm = fmaxf(m1, m2);
            smx[threadIdx.x] = nm;
            ss[threadIdx.x] = s1 * __expf(m1 - nm) + s2 * __expf(m2 - nm);
        }
        __syncthreads();
    }
    if (threadIdx.x == 0) {
        float lse = smx[0] + __logf(ss[0]);
        nll[row] = lse - x[target[row]];
    }
}

// Deterministic final reduction: loss = sum(nll)/B ; avg = loss/T
__global__ __launch_bounds__(256) void final_loss(const float* __restrict__ nll,
                                                  int TB, int B, int T,
                                                  float* __restrict__ out) {
    __shared__ float sb[256];
    float acc = 0.0f;
    for (int i = threadIdx.x; i < TB; i += 256) acc += nll[i];
    sb[threadIdx.x] = acc;
    __syncthreads();
    for (int off = 128; off > 0; off >>= 1) {
        if (threadIdx.x < off) sb[threadIdx.x] += sb[threadIdx.x + off];
        __syncthreads();
    }
    if (threadIdx.x == 0) {
        float loss = sb[0] / (float)B;
        out[0] = loss;
        out[1] = loss / (float)T;
    }
}

// ---------------------------------------------------------------------------
extern "C" void kernel_launch(void* const* d_in, const int* in_sizes, int n_in,
                              void* d_out, int out_size, void* d_ws, size_t ws_size,
                              hipStream_t stream) {
    const int H = 256, B = 32, T_in = 50, T_tgt = 40, T_tag = 20;
    const int V = in_sizes[20];   // out_b length = vocab

    const int*   input_t  = (const int*)d_in[0];
    const int*   target_t = (const int*)d_in[1];
    const int*   tag_t    = (const int*)d_in[2];
    const float* enc_emb  = (const float*)d_in[3];
    const float* enc_Wih  = (const float*)d_in[4];
    const float* enc_Whh  = (const float*)d_in[5];
    const float* enc_bih  = (const float*)d_in[6];
    const float* enc_bhh  = (const float*)d_in[7];
    const float* tag_emb  = (const float*)d_in[8];
    const float* tag_Wih  = (const float*)d_in[9];
    const float* tag_Whh  = (const float*)d_in[10];
    const float* tag_bih  = (const float*)d_in[11];
    const float* tag_bhh  = (const float*)d_in[12];
    const float* proj_W   = (const float*)d_in[13];
    const float* proj_b   = (const float*)d_in[14];
    const float* dec_Wih  = (const float*)d_in[15];
    const float* dec_Whh  = (const float*)d_in[16];
    const float* dec_bih  = (const float*)d_in[17];
    const float* dec_bhh  = (const float*)d_in[18];
    const float* out_W    = (const float*)d_in[19];
    const float* out_b    = (const float*)d_in[20];

    float* ws = (float*)d_ws;
    size_t off = 0;
    float* enc_x  = ws + off; off += (size_t)T_in  * B * H;
    float* tag_x  = ws + off; off += (size_t)T_tag * B * H;
    float* dec_x  = ws + off; off += (size_t)T_tgt * B * H;
    float* enc_gi = ws + off; off += (size_t)T_in  * B * 3 * H;
    float* tag_gi = ws + off; off += (size_t)T_tag * B * 3 * H;
    float* dec_gi = ws + off; off += (size_t)T_tgt * B * 3 * H;
    float* cat_h  = ws + off; off += (size_t)B * 2 * H;   // [tag_h | enc_h]
    float* merged = ws + off; off += (size_t)B * H;
    float* dec_o  = ws + off; off += (size_t)T_tgt * B * H;
    float* nll    = ws + off; off += (size_t)T_tgt * B;

    float* logits = (float*)d_out + 2;   // decoder_outputs region (T,1,B,V)

    auto cdiv = [](int a, int b) { return (a + b - 1) / b; };

    // 1. embedding gathers
    {
        int tot = T_in * B * H;
        gather_emb<<<cdiv(tot, 256), 256, 0, stream>>>(input_t, enc_emb, enc_x, T_in, B, H, 0, 0);
        tot = T_tag * B * H;
        gather_emb<<<cdiv(tot, 256), 256, 0, stream>>>(tag_t, tag_emb, tag_x, T_tag, B, H, 0, 0);
        tot = T_tgt * B * H;
        gather_emb<<<cdiv(tot, 256), 256, 0, stream>>>(target_t, enc_emb, dec_x, T_tgt, B, H, 1, 1);
    }

    // 2. input projections gi = X @ Wih^T + bih  (parallel over time)
    gemm_nt<<<dim3(cdiv(3 * H, 256), cdiv(T_in * B, 32)),  256, 0, stream>>>(
        enc_x, H, enc_Wih, enc_bih, enc_gi, 3 * H, T_in * B,  3 * H, H);
    gemm_nt<<<dim3(cdiv(3 * H, 256), cdiv(T_tag * B, 32)), 256, 0, stream>>>(
        tag_x, H, tag_Wih, tag_bih, tag_gi, 3 * H, T_tag * B, 3 * H, H);
    gemm_nt<<<dim3(cdiv(3 * H, 256), cdiv(T_tgt * B, 32)), 256, 0, stream>>>(
        dec_x, H, dec_Wih, dec_bih, dec_gi, 3 * H, T_tgt * B, 3 * H, H);

    // 3. encoder + tag GRUs run concurrently (grid = 2)
    GruArgs ga = { enc_gi, enc_Whh, enc_bhh, nullptr, nullptr, cat_h + H, 2 * H, T_in  };
    GruArgs gb = { tag_gi, tag_Whh, tag_bhh, nullptr, nullptr, cat_h,     2 * H, T_tag };
    gru_seq<<<2, 1024, 0, stream>>>(ga, gb);

    // 4. merged = [tag_h | enc_h] @ proj_W^T + proj_b
    gemm_nt<<<dim3(cdiv(H, 256), 1), 256, 0, stream>>>(
        cat_h, 2 * H, proj_W, proj_b, merged, H, B, H, 2 * H);

    // 5. decoder GRU
    GruArgs gd = { dec_gi, dec_Whh, dec_bhh, merged, dec_o, nullptr, 0, T_tgt };
    gru_seq<<<1, 1024, 0, stream>>>(gd, gd);

    // 6. logits = dec_o @ out_W^T + out_b  -> straight into d_out
    gemm_nt<<<dim3(cdiv(V, 256), cdiv(T_tgt * B, 32)), 256, 0, stream>>>(
        dec_o, H, out_W, out_b, logits, V, T_tgt * B, V, H);

    // 7. loss
    row_nll<<<T_tgt * B, 256, 0, stream>>>(logits, target_t, V, nll);
    final_loss<<<1, 256, 0, stream>>>(nll, T_tgt * B, B, T_tgt, (float*)d_out);
}